// Aug_MLP_P_RPY_sincos_unc_indep_31756988187355
// MI455X (gfx1250) — compile-verified
//
#include <hip/hip_runtime.h>
#include <math.h>

// MI455X / gfx1250, wave32. fp32 semantics kept via V_WMMA_F32_16X16X4_F32.
// Augmented-batch layout: sample s occupies rows 8s..8s+7 (row 8s = value,
// row 8s+k = d/dq_k tangent). In the 16x16 WMMA C layout, accumulator VGPR 0
// holds the value rows (M=0 and M=8) and VGPR v holds tangent v for both
// samples at the same lane -> sigmoid' chain-rule scaling is lane-local.

typedef __attribute__((ext_vector_type(2))) float v2f;
typedef __attribute__((ext_vector_type(8))) float v8f;

// Hardware transcendental sigmoid: v_exp_f32 + v_rcp_f32 (instead of the
// ~30-op precise expf + Newton-divide sequence). Correct saturation at +/-inf.
__device__ __forceinline__ float fast_sigmoid(float z) {
    return __builtin_amdgcn_rcpf(1.f + __expf(-z));
}

template<int H, int OUT, int SPB, int MODE>
__global__ __launch_bounds__(128)
void fused_mlp_jac(const float* __restrict__ x,
                   const float* __restrict__ W1, const float* __restrict__ b1,
                   const float* __restrict__ W2, const float* __restrict__ b2,
                   const float* __restrict__ W3, const float* __restrict__ b3,
                   float* __restrict__ dout, int Mtot)
{
    constexpr int ROWS    = SPB * 8;       // augmented rows per block
    constexpr int LDA     = H + 2;         // LDS row stride (bank-conflict pad, keeps 8B align)
    constexpr int NSTRIPS = ROWS / 16;     // 16-row WMMA strips
    constexpr int NWAVES  = 4;
    constexpr int WPS     = NWAVES / NSTRIPS;   // waves cooperating per strip (split N)
    constexpr int NT      = H / 16;             // N tiles
    constexpr int NTPW    = NT / WPS;

    extern __shared__ float smem[];
    float* A1 = smem;                   // ROWS * LDA  (layer-1 augmented activations)
    float* A2 = A1 + ROWS * LDA;        // ROWS * LDA  (layer-2 augmented activations)
    float* Y  = A2 + ROWS * LDA;        // ROWS * OUT  (layer-3 raw outputs)

    const int tid = threadIdx.x;
    const int m0  = blockIdx.x * SPB;

    // ---------------- Layer 1: 7 -> H (VALU; tiny) ----------------
    // value: h = sigmoid(q.W1 + b1); tangent k: sigma'(z) * W1[k, j]
    for (int idx = tid; idx < SPB * H; idx += 128) {
        const int s = idx / H, j = idx - s * H;
        float t[7];
        #pragma unroll
        for (int k = 0; k < 7; ++k) t[k] = W1[k * H + j];
        float h = 0.f, d = 0.f;
        if (m0 + s < Mtot) {
            const float* xr = x + (size_t)(m0 + s) * 14;
            float z = b1[j];
            #pragma unroll
            for (int k = 0; k < 7; ++k) z = fmaf(xr[k], t[k], z);
            h = fast_sigmoid(z);
            d = h * (1.f - h);
        }
        float* ar = A1 + (8 * s) * LDA + j;
        ar[0] = h;
        #pragma unroll
        for (int k = 0; k < 7; ++k) ar[(k + 1) * LDA] = d * t[k];
    }
    __syncthreads();

    // ---------------- Layer 2: H -> H GEMM on WMMA f32 16x16x4 ----------------
    const int wave  = tid >> 5;
    const int lane  = tid & 31;
    const int lm    = lane & 15;       // N (and M-row within strip for A frags)
    const int hi    = lane >> 4;       // half-wave select
    const int strip = wave / WPS;
    const int nsub  = wave - strip * WPS;
    const int r0    = strip * 16;      // first augmented row of this wave's strip

    for (int nt = nsub * NTPW; nt < nsub * NTPW + NTPW; ++nt) {
        const int n0 = nt * 16;
        v8f acc = {};
        acc[0] = b2[n0 + lm];          // bias only on the value rows (M=0, M=8)

        // A frag: lane lm holds row r0+lm, cols {k+2hi, k+1+2hi}  -> ds_load_b64
        // B frag: rows k+2hi, k+1+2hi of W2, col n0+lm (L2-resident, shared by all WGs)
        const float* ap = A1 + (r0 + lm) * LDA + 2 * hi;
        const float* bp = W2 + (size_t)(2 * hi) * H + n0 + lm;
        #pragma unroll 8
        for (int k = 0; k < H; k += 4) {
            v2f a = *(const v2f*)(ap + k);
            v2f b;
            b.x = bp[(size_t)k * H];
            b.y = bp[(size_t)k * H + H];
            acc = __builtin_amdgcn_wmma_f32_16x16x4_f32(
                      false, a, false, b, (short)0, acc, false, false);
        }

        // Fused activation + forward-mode chain rule, lane-local on accumulators:
        float sv = fast_sigmoid(acc[0]);
        float dv = sv * (1.f - sv);
        acc[0] = sv;
        #pragma unroll
        for (int v = 1; v < 8; ++v) acc[v] *= dv;

        float* a2 = A2 + (r0 + 8 * hi) * LDA + n0 + lm;
        #pragma unroll
        for (int v = 0; v < 8; ++v) a2[v * LDA] = acc[v];
    }
    __syncthreads();

    // ---------------- Layer 3: H -> OUT (VALU; tiny) ----------------
    for (int idx = tid; idx < ROWS * OUT; idx += 128) {
        const int o = idx % OUT, row = idx / OUT;
        const float* ar = A2 + row * LDA;
        float acc3 = (row & 7) ? 0.f : b3[o];   // bias only on value rows
        #pragma unroll 8
        for (int j = 0; j < H; ++j) acc3 = fmaf(ar[j], W3[j * OUT + o], acc3);
        Y[row * OUT + o] = acc3;
    }
    __syncthreads();

    // ---------------- Heads + output scatter ----------------
    const size_t offJ  = (size_t)Mtot * 18;
    const size_t offOA = offJ + (size_t)Mtot * 63;
    const size_t offJA = offOA + (size_t)Mtot * 6;

    for (int s = tid; s < SPB; s += 128) {
        const size_t m = (size_t)m0 + s;
        if (m >= (size_t)Mtot) continue;
        const float* qd = x + m * 14 + 7;

        if (MODE == 0) {                       // pos net: P, J_P, vel
            #pragma unroll
            for (int o = 0; o < 3; ++o) {
                float P = Y[(8 * s) * 3 + o];
                float Jr[7]; float vel = 0.f;
                #pragma unroll
                for (int k = 0; k < 7; ++k) {
                    Jr[k] = Y[(8 * s + 1 + k) * 3 + o];
                    vel = fmaf(Jr[k], qd[k], vel);
                }
                dout[m * 18 + o]            = P;
                dout[m * 18 + 9 + o]        = vel;
                dout[offOA + m * 6 + o]     = P;
                #pragma unroll
                for (int k = 0; k < 7; ++k) {
                    dout[offJ  + m * 63 + o * 7 + k] = Jr[k];
                    dout[offJA + m * 42 + o * 7 + k] = Jr[k];
                }
            }
        } else {                               // rpy head r = MODE-1
            const int r = MODE - 1;
            float y0 = Y[(8 * s) * 2 + 0], y1 = Y[(8 * s) * 2 + 1];
            float sv = sinf(y0), cv = cosf(y1);
            float c0 = cosf(y0), s1 = sinf(y1);
            float ang = atan2f(sv, cv);
            float den = sv * sv + cv * cv;
            float v0 = 0.f, v1 = 0.f;
            float J0[7], J1[7], J2[7];
            #pragma unroll
            for (int k = 0; k < 7; ++k) {
                float jy0 = Y[(8 * s + 1 + k) * 2 + 0];
                float jy1 = Y[(8 * s + 1 + k) * 2 + 1];
                J0[k] = c0 * jy0;              // d sin(y0)/dq
                J1[k] = -s1 * jy1;             // d cos(y1)/dq
                J2[k] = (cv * J0[k] - sv * J1[k]) / den;   // d atan2(s,c)/dq
                v0 = fmaf(J0[k], qd[k], v0);
                v1 = fmaf(J1[k], qd[k], v1);
            }
            dout[m * 18 + 3  + r] = sv;
            dout[m * 18 + 6  + r] = cv;
            dout[m * 18 + 12 + r] = v0;
            dout[m * 18 + 15 + r] = v1;
            dout[offOA + m * 6 + 3 + r] = ang;
            #pragma unroll
            for (int k = 0; k < 7; ++k) {
                dout[offJ  + m * 63 + (3 + r) * 7 + k] = J0[k];
                dout[offJ  + m * 63 + (6 + r) * 7 + k] = J1[k];
                dout[offJA + m * 42 + (3 + r) * 7 + k] = J2[k];
            }
        }
    }
}

extern "C" void kernel_launch(void* const* d_in, const int* in_sizes, int n_in,
                              void* d_out, int out_size, void* d_ws, size_t ws_size,
                              hipStream_t stream)
{
    (void)n_in; (void)out_size; (void)d_ws; (void)ws_size;
    const float* x = (const float*)d_in[0];
    float* out = (float*)d_out;
    const int Mtot = in_sizes[0] / 14;

    // inputs: 1..6 pos W0,b0,W1,b1,W2,b2 ; 7..12 roll ; 13..18 pitch ; 19..24 yaw
    auto f = [&](int i) { return (const float*)d_in[i]; };

    const size_t sh_pos = (size_t)(2 * 32 * (512 + 2) + 32 * 3) * sizeof(float); // ~132 KB
    const size_t sh_rpy = (size_t)(2 * 64 * (256 + 2) + 64 * 2) * sizeof(float); // ~132 KB
    dim3 blk(128);
    const int gp = (Mtot + 3) / 4;   // 4 samples (32 augmented rows) per block
    const int gr = (Mtot + 7) / 8;   // 8 samples (64 augmented rows) per block

    fused_mlp_jac<512, 3, 4, 0><<<gp, blk, sh_pos, stream>>>(
        x, f(1), f(2), f(3), f(4), f(5), f(6), out, Mtot);
    fused_mlp_jac<256, 2, 8, 1><<<gr, blk, sh_rpy, stream>>>(
        x, f(7), f(8), f(9), f(10), f(11), f(12), out, Mtot);
    fused_mlp_jac<256, 2, 8, 2><<<gr, blk, sh_rpy, stream>>>(
        x, f(13), f(14), f(15), f(16), f(17), f(18), out, Mtot);
    fused_mlp_jac<256, 2, 8, 3><<<gr, blk, sh_rpy, stream>>>(
        x, f(19), f(20), f(21), f(22), f(23), f(24), out, Mtot);
}